// ROIPool_37177236914178
// MI455X (gfx1250) — compile-verified
//
#include <hip/hip_runtime.h>
#include <math.h>

typedef __attribute__((ext_vector_type(2))) float v2f;
typedef __attribute__((ext_vector_type(4))) float v4f;
typedef __attribute__((ext_vector_type(8))) float v8f;
typedef __attribute__((ext_vector_type(4))) int   v4i;

#define NBOX 128
#define C_CH 256
#define HW   26
#define KDIM 2304   // 256*9
#define N1   1024
#define N2   512

#define KC         64              // K-chunk staged in LDS (divides 2304 and 1024)
#define MTILES     4               // 16-row tiles per wave
#define MBLK       (MTILES * 16)   // 64 rows of A shared by the whole block
#define LDS_STRIDE 68              // padded row stride (floats): 272B = 17*16, bank-spread

#if defined(__gfx1250__) && __has_builtin(__builtin_amdgcn_global_load_async_to_lds_b128)
#define HAVE_ASYNC_LDS 1
#else
#define HAVE_ASYNC_LDS 0
#endif

// ---------------------------------------------------------------------------
// Kernel 1: ROI crop + AdaptiveMaxPool2d(3) -> feats [NBOX, KDIM]
// feats[n, c*9 + i*3 + j]; bins: start=floor(i*L/3), end=ceil((i+1)*L/3).
// ---------------------------------------------------------------------------
__global__ void roi_pool_kernel(const float* __restrict__ x,     // [256,26,26]
                                const int*   __restrict__ det,   // [NBOX,4]
                                float*       __restrict__ feats) // [NBOX,KDIM]
{
    const int n = blockIdx.x;
    const int c = threadIdx.x;

    const int x1 = det[n * 4 + 0];
    const int y1 = det[n * 4 + 1];
    const int x2 = det[n * 4 + 2];
    const int y2 = det[n * 4 + 3];
    const int hl = y2 - y1 + 1;
    const int wl = x2 - x1 + 1;

    const float* __restrict__ xc = x + (size_t)c * (HW * HW);
    float* __restrict__ outp = feats + (size_t)n * KDIM + c * 9;

#pragma unroll
    for (int i = 0; i < 3; ++i) {
        const int hs = y1 + (i * hl) / 3;
        const int he = y1 + ((i + 1) * hl + 2) / 3;
#pragma unroll
        for (int j = 0; j < 3; ++j) {
            const int ws = x1 + (j * wl) / 3;
            const int we = x1 + ((j + 1) * wl + 2) / 3;
            float m = -3.402823466e38f;
            for (int h = hs; h < he; ++h) {
                const float* row = xc + h * HW;
                for (int w = ws; w < we; ++w)
                    m = fmaxf(m, row[w]);
            }
            outp[i * 3 + j] = m;
        }
    }
}

// ---------------------------------------------------------------------------
// GEMM: C[M,N] = A[M,K] @ B[K,N] + bias, exact fp32 via v_wmma_f32_16x16x4_f32.
//
// Block = 256 threads = 8 waves. The block owns 64 rows of A (MBLK) staged in
// LDS via async global->LDS copies (ASYNCcnt) and 8 adjacent 16-col N-tiles,
// one per wave. Each wave register-blocks 4 M-tiles -> 4 WMMAs per B fragment.
//
// Fragment layouts per CDNA5 ISA (f32, wave32):
//   A 16x4 : lane L -> A[m=L&15][k0], A[m][k0+1], k0 = k + 2*(L>=16)
//   B 4x16 : lane L -> B[k0][n=L&15], B[k0+1][n]
//   C/D    : element (vgpr r, lane L) = D[r + 8*(L>=16)][L&15]
// All control flow is wave-uniform -> EXEC all-1s at every WMMA.
// ---------------------------------------------------------------------------
__global__ void wmma_gemm_bias(const float* __restrict__ A,    // [M,K]
                               const float* __restrict__ B,    // [K,N]
                               const float* __restrict__ bias, // [N]
                               float*       __restrict__ C,    // [M,N]
                               int M, int N, int K)
{
    __shared__ float lds_a[MBLK * LDS_STRIDE];   // 64 * 68 * 4B = 17408 B

    const int wave = threadIdx.x >> 5;
    const int lane = threadIdx.x & 31;
    const int lo   = lane & 15;
    const int hi   = lane >> 4;                  // 0 or 1

    const int tiles_n       = N >> 4;
    const int blocks_per_mg = tiles_n >> 3;      // 8 N-tiles per block
    const int mg = blockIdx.x / blocks_per_mg;
    const int ng = blockIdx.x - mg * blocks_per_mg;
    const int tn = ng * 8 + wave;
    const int m0 = mg * MBLK;
    const int n  = tn * 16 + lo;

    // Cooperative A staging: thread t copies 16 floats of row (t>>2), seg (t&3).
    const int crow = threadIdx.x >> 2;
    const int cseg = threadIdx.x & 3;
    const float* __restrict__ gsrc = A + (size_t)(m0 + crow) * K + cseg * 16;
    float* ldst = &lds_a[crow * LDS_STRIDE + cseg * 16];

    v8f acc[MTILES] = {};

    for (int kc = 0; kc < K; kc += KC) {
        __syncthreads();   // previous chunk's compute done before overwrite
#if HAVE_ASYNC_LDS
#pragma unroll
        for (int q = 0; q < 4; ++q) {
            // builtin expects (int4 AS1*, int4 AS3*, imm offset, imm cpol)
            __builtin_amdgcn_global_load_async_to_lds_b128(
                (__attribute__((address_space(1))) v4i*)(gsrc + kc + q * 4),
                (__attribute__((address_space(3))) v4i*)(ldst + q * 4),
                /*offset=*/0, /*cpol=*/0);
        }
#if __has_builtin(__builtin_amdgcn_s_wait_asynccnt)
        __builtin_amdgcn_s_wait_asynccnt(0);
#else
        asm volatile("s_wait_asynccnt 0x0" ::: "memory");
#endif
#else
#pragma unroll
        for (int q = 0; q < 4; ++q) {
            v4f v = *(const v4f*)(gsrc + kc + q * 4);   // global_load_b128
            *(v4f*)(ldst + q * 4) = v;                  // ds_store_b128
        }
#endif
        __syncthreads();

        for (int kk = 0; kk < KC; kk += 4) {
            const int k0g = kc + kk + 2 * hi;
            // B fragment from global: lanes 0..15 read consecutive n -> coalesced
            v2f b;
            b.x = B[(size_t)k0g * N + n];
            b.y = B[(size_t)(k0g + 1) * N + n];
#pragma unroll
            for (int i = 0; i < MTILES; ++i) {
                // A fragment from LDS: ds_load_b64, conflict-free (stride 68)
                v2f a = *(const v2f*)&lds_a[(i * 16 + lo) * LDS_STRIDE + kk + 2 * hi];
                acc[i] = __builtin_amdgcn_wmma_f32_16x16x4_f32(
                             false, a, false, b, (short)0, acc[i], false, false);
            }
        }
    }

    const float bv = bias[n];
#pragma unroll
    for (int i = 0; i < MTILES; ++i) {
#pragma unroll
        for (int r = 0; r < 8; ++r) {
            const int mm = m0 + i * 16 + r + 8 * hi;
            C[(size_t)mm * N + n] = acc[i][r] + bv;     // coalesced per half-wave
        }
    }
}

// ---------------------------------------------------------------------------
// Kernel 4: out[n] = softplus(h2[n,:] . W3 + b3)   (numerically stable)
// ---------------------------------------------------------------------------
__global__ void head_kernel(const float* __restrict__ h2,  // [NBOX, N2]
                            const float* __restrict__ W3,  // [N2]
                            const float* __restrict__ b3,  // [1]
                            float*       __restrict__ out) // [NBOX]
{
    const int n = threadIdx.x;
    if (n >= NBOX) return;
    float s = b3[0];
    const float* __restrict__ row = h2 + (size_t)n * N2;
    for (int k = 0; k < N2; ++k)
        s = fmaf(row[k], W3[k], s);
    out[n] = fmaxf(s, 0.0f) + log1pf(expf(-fabsf(s)));
}

// ---------------------------------------------------------------------------
extern "C" void kernel_launch(void* const* d_in, const int* in_sizes, int n_in,
                              void* d_out, int out_size, void* d_ws, size_t ws_size,
                              hipStream_t stream)
{
    const float* x   = (const float*)d_in[0];   // [1,256,26,26]
    const int*   det = (const int*)  d_in[1];   // [128,4]
    const float* W1  = (const float*)d_in[2];   // [2304,1024]
    const float* b1  = (const float*)d_in[3];   // [1024]
    const float* W2  = (const float*)d_in[4];   // [1024,512]
    const float* b2  = (const float*)d_in[5];   // [512]
    const float* W3  = (const float*)d_in[6];   // [512,1]
    const float* b3  = (const float*)d_in[7];   // [1]
    float* out = (float*)d_out;                 // [128,1]

    float* feats = (float*)d_ws;                // 128*2304
    float* h1    = feats + NBOX * KDIM;         // 128*1024
    float* h2    = h1 + NBOX * N1;              // 128*512

    // 1) ROI adaptive maxpool
    roi_pool_kernel<<<NBOX, C_CH, 0, stream>>>(x, det, feats);

    // 2) feats[128,2304] @ W1 + b1 : (128/64 mg) * (1024/16/8) = 16 blocks
    wmma_gemm_bias<<<16, 256, 0, stream>>>(feats, W1, b1, h1, NBOX, N1, KDIM);

    // 3) h1[128,1024] @ W2 + b2    : 2 * (512/16/8) = 8 blocks
    wmma_gemm_bias<<<8, 256, 0, stream>>>(h1, W2, b2, h2, NBOX, N2, N1);

    // 4) final dot + softplus
    head_kernel<<<1, NBOX, 0, stream>>>(h2, W3, b3, out);
}